// GraphEncoder_70755291234309
// MI455X (gfx1250) — compile-verified
//
#include <hip/hip_runtime.h>
#include <hip/hip_bf16.h>

typedef __attribute__((ext_vector_type(2))) float v2f;
typedef __attribute__((ext_vector_type(8))) float v8f;

#define NINP 64

__device__ __forceinline__ void atomAddF(float* p, float v) {
    __hip_atomic_fetch_add(p, v, __ATOMIC_RELAXED, __HIP_MEMORY_SCOPE_AGENT);
}

// ---------------------------------------------------------------- degree / norm
__global__ void k_init_deg(float* __restrict__ deg, int n) {
    int i = blockIdx.x * blockDim.x + threadIdx.x;
    if (i < n) deg[i] = 1.0f;                      // self-loop contribution
}

__global__ void k_accum_deg(const int* __restrict__ dst, float* __restrict__ deg, int e) {
    int i = blockIdx.x * blockDim.x + threadIdx.x;
    if (i < e) atomAddF(&deg[dst[i]], 1.0f);
}

__global__ void k_finalize_dinv(float* __restrict__ deg, int n) {
    int i = blockIdx.x * blockDim.x + threadIdx.x;
    if (i < n) deg[i] = rsqrtf(deg[i]);            // deg >= 1 always
}

__global__ void k_edge_norm(const int* __restrict__ src, const int* __restrict__ dst,
                            const float* __restrict__ dinv, float* __restrict__ norm, int e) {
    int i = blockIdx.x * blockDim.x + threadIdx.x;
    if (i < e) norm[i] = dinv[src[i]] * dinv[dst[i]];
}

// ---------------------------------------------------------------- WMMA GEMM
// Y[nrows, NCOLS] = X[nrows, K] @ W[K, NCOLS]; one wave per 16x16 tile.
// f32 A frag: lanes 0-15 -> K = k0+0,k0+1 ; lanes 16-31 -> K = k0+2,k0+3
// f32 B frag: lanes 0-15 -> N=lane, same K split by lane-half.
template <int K, int NCOLS>
__global__ void k_gemm_wmma(const float* __restrict__ X, const float* __restrict__ W,
                            float* __restrict__ Y, int nrows) {
    constexpr int NT = NCOLS / 16;
    const int lane = threadIdx.x & 31;
    const int wib  = threadIdx.x >> 5;
    const long long wave = (long long)blockIdx.x * (blockDim.x >> 5) + wib;
    const int tiles_m = (nrows + 15) >> 4;
    if (wave >= (long long)tiles_m * NT) return;
    const int tm = (int)(wave / NT);
    const int tn = (int)(wave % NT);

    const int half = lane >> 4;          // 0 | 1
    const int l16  = lane & 15;
    int rowA = tm * 16 + l16;
    if (rowA >= nrows) rowA = nrows - 1; // clamp, EXEC stays all-ones
    const int colB = tn * 16 + l16;

    const float* __restrict__ arow = X + (long long)rowA * K;
    v8f acc = {};
#pragma unroll
    for (int k0 = 0; k0 < K; k0 += 4) {
        const int ka = k0 + half * 2;
        v2f a, b;
        a.x = arow[ka];
        a.y = arow[ka + 1];
        b.x = W[(long long)ka * NCOLS + colB];
        b.y = W[(long long)(ka + 1) * NCOLS + colB];
        acc = __builtin_amdgcn_wmma_f32_16x16x4_f32(
            /*neg_a=*/false, a, /*neg_b=*/false, b,
            /*c_mod=*/(short)0, acc, /*reuse_a=*/false, /*reuse_b=*/false);
    }
    // C/D layout: acc[r] holds element (m = r + 8*half, n = l16)
#pragma unroll
    for (int r = 0; r < 8; ++r) {
        const int m = tm * 16 + r + 8 * half;
        if (m < nrows) Y[(long long)m * NCOLS + tn * 16 + l16] = acc[r];
    }
}

// ---------------------------------------------------------------- aggregation
// OUT[i][c] = dinv[i]^2 * H[i][c] + bias[c]   (self-loop term + bias init)
template <int C>
__global__ void k_selfloop_init(const float* __restrict__ H, const float* __restrict__ dinv,
                                const float* __restrict__ bias, float* __restrict__ OUT,
                                int n) {
    long long tid = (long long)blockIdx.x * blockDim.x + threadIdx.x;
    if (tid >= (long long)n * C) return;
    const int i = (int)(tid / C);
    const int c = (int)(tid % C);
    const float di = dinv[i];
    OUT[tid] = di * di * H[tid] + bias[c];
}

// OUT[dst] += norm[e] * H[src]  — 4 channels per thread, native f32 atomics
template <int C>
__global__ void k_edge_scatter(const int* __restrict__ src, const int* __restrict__ dst,
                               const float* __restrict__ norm, const float* __restrict__ H,
                               float* __restrict__ OUT, int e) {
    constexpr int CH = C / 4;
    long long tid = (long long)blockIdx.x * blockDim.x + threadIdx.x;
    if (tid >= (long long)e * CH) return;
    const int ei = (int)(tid / CH);
    const int c  = (int)(tid % CH) * 4;
    const int s = src[ei], d = dst[ei];
    const float w = norm[ei];
    const float4 h4 = *(const float4*)(H + (long long)s * C + c);
    float* op = OUT + (long long)d * C + c;
    atomAddF(op + 0, w * h4.x);
    atomAddF(op + 1, w * h4.y);
    atomAddF(op + 2, w * h4.z);
    atomAddF(op + 3, w * h4.w);
}

// ---------------------------------------------------------------- final gather
__global__ void k_gather(const int* __restrict__ tok, const float* __restrict__ table,
                         float* __restrict__ out, int ntok) {
    long long tid = (long long)blockIdx.x * blockDim.x + threadIdx.x;
    if (tid >= (long long)ntok * NINP) return;
    const int t = (int)(tid >> 6);
    const int c = (int)(tid & 63);
    out[tid] = table[(long long)tok[t] * NINP + c];
}

// ---------------------------------------------------------------- launch
static inline int gridFor(long long work, int block) {
    return (int)((work + block - 1) / block);
}

extern "C" void kernel_launch(void* const* d_in, const int* in_sizes, int n_in,
                              void* d_out, int out_size, void* d_ws, size_t ws_size,
                              hipStream_t stream) {
    const int*   tokens = (const int*)d_in[0];
    const int*   eidx   = (const int*)d_in[1];
    const float* emb    = (const float*)d_in[2];
    const float* W1     = (const float*)d_in[3];
    const float* b1     = (const float*)d_in[4];
    const float* W2     = (const float*)d_in[5];
    const float* b2     = (const float*)d_in[6];
    float* out = (float*)d_out;

    const int T = in_sizes[0];           // B*L tokens
    const int E = in_sizes[1] / 2;       // edges
    const int N = in_sizes[2] / NINP;    // nodes

    const int* src = eidx;
    const int* dst = eidx + E;

    // workspace carve-up (floats), 256B-aligned blocks; H2 aliases H, TABLE aliases AGG
    auto alignUp = [](long long x) { return (x + 63) & ~63LL; };
    float* ws   = (float*)d_ws;
    float* dinv = ws;                                 long long o = alignUp(N);
    float* norm = ws + o;                             o += alignUp(E);
    float* H    = ws + o;                             o += (long long)N * 2 * NINP;
    float* AGG  = ws + o;
    float* H2    = H;    // [N,64], dead H overwritten after AGG computed
    float* TABLE = AGG;  // [N,64], dead AGG overwritten after H2 computed... (see order below)
    (void)ws_size; (void)n_in; (void)out_size;

    const int BLK = 256;

    // 1) degree -> dinv, per-edge norm (shared by both layers)
    k_init_deg<<<gridFor(N, BLK), BLK, 0, stream>>>(dinv, N);
    k_accum_deg<<<gridFor(E, BLK), BLK, 0, stream>>>(dst, dinv, E);
    k_finalize_dinv<<<gridFor(N, BLK), BLK, 0, stream>>>(dinv, N);
    k_edge_norm<<<gridFor(E, BLK), BLK, 0, stream>>>(src, dst, dinv, norm, E);

    // 2) layer 1: H = emb @ W1  (WMMA f32 16x16x4), then normalized scatter-add
    {
        const long long waves = (long long)((N + 15) / 16) * (2 * NINP / 16);
        k_gemm_wmma<NINP, 2 * NINP><<<gridFor(waves * 32, BLK), BLK, 0, stream>>>(emb, W1, H, N);
        k_selfloop_init<2 * NINP><<<gridFor((long long)N * 2 * NINP, BLK), BLK, 0, stream>>>(H, dinv, b1, AGG, N);
        k_edge_scatter<2 * NINP><<<gridFor((long long)E * (2 * NINP / 4), BLK), BLK, 0, stream>>>(src, dst, norm, H, AGG, E);
    }

    // 3) layer 2: H2 = AGG @ W2 (H2 aliases H — H is dead), then scatter into TABLE (aliases AGG
    //    only AFTER H2 is fully computed; k_selfloop_init reads dead-H2's source AGG... careful:
    //    TABLE aliases AGG, but k_selfloop_init<64> reads H2 and writes TABLE=AGG — AGG is dead
    //    once the GEMM consumed it, so this is safe with kernel-order serialization on `stream`.
    {
        const long long waves = (long long)((N + 15) / 16) * (NINP / 16);
        k_gemm_wmma<2 * NINP, NINP><<<gridFor(waves * 32, BLK), BLK, 0, stream>>>(AGG, W2, H2, N);
        k_selfloop_init<NINP><<<gridFor((long long)N * NINP, BLK), BLK, 0, stream>>>(H2, dinv, b2, TABLE, N);
        k_edge_scatter<NINP><<<gridFor((long long)E * (NINP / 4), BLK), BLK, 0, stream>>>(src, dst, norm, H2, TABLE, E);
    }

    // 4) out[b,l,:] = TABLE[token]
    k_gather<<<gridFor((long long)T * NINP, BLK), BLK, 0, stream>>>(tokens, TABLE, out, T);
}